// FastFeedForward_21406117003532
// MI455X (gfx1250) — compile-verified
//
#include <hip/hip_runtime.h>

#define F_DIM   1024
#define TILE    16
#define THREADS 256
#define NWAVES  8
#define DEPTH   10

typedef __attribute__((ext_vector_type(2))) float v2f;
typedef __attribute__((ext_vector_type(8))) float v8f;

__global__ __launch_bounds__(THREADS)
void fff_tree_wmma_kernel(const float* __restrict__ xin,
                          const float* __restrict__ Xw,
                          const float* __restrict__ Yw,
                          float* __restrict__ out)
{
    __shared__ float lamPart[NWAVES][TILE];
    __shared__ float lamLds[TILE];
    __shared__ int   curBuf[2][TILE];        // double-buffered tree node per sample

    const int t    = threadIdx.x;
    const int lane = t & 31;
    const int wv   = t >> 5;                 // wave id 0..7 -> K slice [wv*128, wv*128+128)
    const long sample0 = (long)blockIdx.x * TILE;

    // ---- WMMA A/B lane mapping (ISA 7.12.2, 32-bit 16x4 A matrix) ----
    const int mrow  = lane & 15;             // A row (= sample), B col (= sample)
    const int koff  = (lane >> 4) << 1;      // lanes 0-15 -> K {0,1}; lanes 16-31 -> K {2,3}
    const int kbase = wv << 7;               // this wave's K offset

    // ---- load x tile straight into A-operand register layout (reused 10x) ----
    v2f xa[32];
    {
        const float* xrow = xin + (sample0 + mrow) * F_DIM;
        #pragma unroll
        for (int ch = 0; ch < 32; ++ch)
            xa[ch] = *(const v2f*)(xrow + kbase + (ch << 2) + koff);
    }

    if (t < TILE) curBuf[0][t] = 0;
    __syncthreads();

    // ---- per-thread output accumulator: sample s, feature group g ----
    const int s = t >> 4;                    // 0..15
    const int g = t & 15;                    // 0..15
    float4 yacc[16];
    #pragma unroll
    for (int c = 0; c < 16; ++c) yacc[c] = make_float4(0.f, 0.f, 0.f, 0.f);

    #pragma unroll 1
    for (int d = 0; d < DEPTH; ++d) {
        const int p = d & 1;

        // ===== phase 1: batched dots via v_wmma_f32_16x16x4_f32, dual accumulators =====
        const int    cn   = curBuf[p][mrow];             // node feeding this lane's B column
        const float* brow = Xw + (long)cn * F_DIM;
        v8f acc0 = {0.f, 0.f, 0.f, 0.f, 0.f, 0.f, 0.f, 0.f};
        v8f acc1 = {0.f, 0.f, 0.f, 0.f, 0.f, 0.f, 0.f, 0.f};
        #pragma unroll
        for (int ch = 0; ch < 32; ch += 2) {
            const int k0 = kbase + (ch << 2);
            v2f b0 = *(const v2f*)(brow + k0 + koff);         // B[k,n] = X[cur_n, k]
            v2f b1 = *(const v2f*)(brow + k0 + 4 + koff);
            acc0 = __builtin_amdgcn_wmma_f32_16x16x4_f32(
                false, xa[ch],     false, b0, (short)0, acc0, false, false);
            acc1 = __builtin_amdgcn_wmma_f32_16x16x4_f32(
                false, xa[ch + 1], false, b1, (short)0, acc1, false, false);
        }
        // diag extraction: D[m,m].  m<8 -> VGPR m @ lane m ; m=8+j -> VGPR j @ lane 24+j
        {
            const int j = (lane < 8) ? lane : (lane - 24);
            float pick = 0.f;
            #pragma unroll
            for (int e = 0; e < 8; ++e) if (j == e) pick = acc0[e] + acc1[e];
            if (lane < 8)        lamPart[wv][lane]      = pick;
            else if (lane >= 24) lamPart[wv][lane - 16] = pick;
        }
        __syncthreads();                                  // b1

        // ===== phase 2: reduce 8 wave partials; compute next node (double buffer) =====
        if (t < TILE) {
            float l = 0.f;
            #pragma unroll
            for (int w = 0; w < NWAVES; ++w) l += lamPart[w][t];
            lamLds[t] = l;
            curBuf[p ^ 1][t] = curBuf[p][t] * 2 + 1 + (l > 0.f ? 1 : 0);
        }
        __syncthreads();                                  // b2

        // ===== phase 3: y += lam * Y[cur] ; prefetch exact next-level rows =====
        const int   cs = curBuf[p][s];
        const float ls = lamLds[s];

        if (d < DEPTH - 1) {
            const int nc = curBuf[p ^ 1][s];              // exact next node, already known
            const float* xn = Xw + (long)nc * F_DIM;
            const float* yn = Yw + (long)nc * F_DIM;
            __builtin_prefetch(xn + (g << 6),      0, 0); // 16 threads x 256B = full 4KB row
            __builtin_prefetch(xn + (g << 6) + 32, 0, 0); // cover alternate 128B lines
            __builtin_prefetch(yn + (g << 6),      0, 0);
            __builtin_prefetch(yn + (g << 6) + 32, 0, 0);
        }

        const float4* yrow = (const float4*)(Yw + (long)cs * F_DIM);
        #pragma unroll
        for (int c = 0; c < 16; ++c) {
            float4 yv = yrow[(c << 4) + g];               // feature c*64 + g*4
            yacc[c].x += ls * yv.x;
            yacc[c].y += ls * yv.y;
            yacc[c].z += ls * yv.z;
            yacc[c].w += ls * yv.w;
        }
        // no trailing barrier: b1 of the next depth protects lamLds/curBuf reuse
    }

    // ---- coalesced store ----
    float4* orow = (float4*)(out + (sample0 + s) * F_DIM);
    #pragma unroll
    for (int c = 0; c < 16; ++c) orow[(c << 4) + g] = yacc[c];
}

extern "C" void kernel_launch(void* const* d_in, const int* in_sizes, int n_in,
                              void* d_out, int out_size, void* d_ws, size_t ws_size,
                              hipStream_t stream) {
    const float* x  = (const float*)d_in[0];   // oldx  [8,4096,1024] f32
    const float* Xw = (const float*)d_in[1];   // X     [1023,1024]   f32
    const float* Yw = (const float*)d_in[2];   // Y     [1023,1024]   f32
    float* out = (float*)d_out;

    const int B = in_sizes[0] / F_DIM;         // 32768 samples
    const int blocks = B / TILE;               // 2048 workgroups of 256 threads

    fff_tree_wmma_kernel<<<blocks, THREADS, 0, stream>>>(x, Xw, Yw, out);
}